// Scatter_Diffuse_Second_79869211836605
// MI455X (gfx1250) — compile-verified
//
#include <hip/hip_runtime.h>
#include <stdint.h>

// ---------------------------------------------------------------------------
// Graph scattering transform for MI455X (gfx1250).
//  * CSR-by-destination built per launch (count -> scan -> scatter).
//  * Diffusion = gather-SpMM; edge stream staged into LDS via CDNA5 async
//    loads (global_load_async_to_lds_b64 / s_wait_asynccnt, ASYNCcnt path).
//  * Feature states (12.8-51.2 MB) are L2-resident (192 MB L2) -> gathers are
//    L2-bound; HBM traffic ~= the 934 MB output write (~41 us at 23.3 TB/s).
//  * First-cascade epilogue (wavelet differences) is a real 4x4 x 4xC dense
//    matmul per 16-node tile -> executed on V_WMMA_F32_16X16X4_F32 in full
//    f32 precision (a16 term folded at store; C accumulator = 0).
// ---------------------------------------------------------------------------

typedef __attribute__((ext_vector_type(2))) float v2f;
typedef __attribute__((ext_vector_type(8))) float v8f;

#define EDGE_CHUNK 512   // LDS staging ring: 512 x 8B = 4 KB

__global__ void __launch_bounds__(256)
zero2_kernel(int* __restrict__ a, int* __restrict__ b, int n) {
  int i = blockIdx.x * 256 + threadIdx.x;
  if (i < n) { a[i] = 0; b[i] = 0; }
}

__global__ void __launch_bounds__(256)
count_kernel(const int* __restrict__ col, int* __restrict__ deg, int e) {
  int i = blockIdx.x * 256 + threadIdx.x;
  if (i < e) atomicAdd(&deg[col[i]], 1);
}

__global__ void __launch_bounds__(256)
dinv_kernel(const int* __restrict__ deg, float* __restrict__ dinv, int n) {
  int i = blockIdx.x * 256 + threadIdx.x;
  if (i < n) {
    int d = deg[i];
    dinv[i] = (d > 0) ? rsqrtf((float)d) : 0.0f;   // deg^-0.5, inf -> 0
  }
}

// Single-workgroup exclusive scan over deg[0..n-1] -> ptr[0..n]  (n = 50k: cheap)
__global__ void __launch_bounds__(1024)
scan_kernel(const int* __restrict__ deg, int* __restrict__ ptr, int n) {
  __shared__ int buf[1024];
  __shared__ int carry;
  const int tid = threadIdx.x;
  if (tid == 0) carry = 0;
  __syncthreads();
  for (int base = 0; base < n; base += 1024) {
    int i = base + tid;
    int v = (i < n) ? deg[i] : 0;
    buf[tid] = v;
    __syncthreads();
    for (int off = 1; off < 1024; off <<= 1) {
      int tv = (tid >= off) ? buf[tid - off] : 0;
      __syncthreads();
      buf[tid] += tv;
      __syncthreads();
    }
    int c = carry;
    if (i < n) ptr[i] = c + buf[tid] - v;   // exclusive
    __syncthreads();
    if (tid == 0) carry = c + buf[1023];
    __syncthreads();
  }
  if (tid == 0) ptr[n] = carry;
}

// Scatter edges into CSR order: edata[pos] = { src_row, norm = dinv[row]*dinv[col] }
__global__ void __launch_bounds__(256)
scatter_kernel(const int* __restrict__ row, const int* __restrict__ col,
               const float* __restrict__ dinv, const int* __restrict__ ptr,
               int* __restrict__ fill, int2* __restrict__ edata, int e) {
  int i = blockIdx.x * 256 + threadIdx.x;
  if (i < e) {
    int r = row[i], c = col[i];
    float w = dinv[r] * dinv[c];
    int pos = ptr[c] + atomicAdd(&fill[c], 1);
    edata[pos] = make_int2(r, __float_as_int(w));
  }
}

// ---------------------------------------------------------------------------
// diffuse: hout = 0.5*(h + S h). One block owns TILE dest nodes; their CSR
// edge range is contiguous and staged into LDS in 4 KB chunks with CDNA5
// async loads. Each F/4-thread group accumulates float4 (contiguous 256B-1KB
// reads of h[src] per edge). If out2 != null (second cascade) the result row
// 5 + k*17 + t of the [N,73,64] output is written directly (k-major layout).
// ---------------------------------------------------------------------------
template <int F>
__global__ void __launch_bounds__(256)
diffuse_kernel(const float* __restrict__ h, float* __restrict__ hout,
               const int* __restrict__ ptr, const int2* __restrict__ edata,
               int n, float* __restrict__ out2, int t) {
  constexpr int GROUP = F / 4;       // threads per destination node
  constexpr int TILE  = 256 / GROUP; // destination nodes per block
  __shared__ int2 ebuf[EDGE_CHUNK];

  const int tid  = threadIdx.x;
  const int base = blockIdx.x * TILE;
  const int node = base + tid / GROUP;
  const int fo   = (tid % GROUP) * 4;

  const int tileEndNode = (base + TILE < n) ? (base + TILE) : n;
  const int eTileBeg = ptr[base];
  const int eTileEnd = ptr[tileEndNode];

  int myBeg = 0, myEnd = 0;
  if (node < n) { myBeg = ptr[node]; myEnd = ptr[node + 1]; }

  float4 acc = make_float4(0.f, 0.f, 0.f, 0.f);

  for (int cb = eTileBeg; cb < eTileEnd; cb += EDGE_CHUNK) {
    const int cnt = (eTileEnd - cb < EDGE_CHUNK) ? (eTileEnd - cb) : EDGE_CHUNK;

    // Async-stage this chunk of {src,norm} records into LDS (ASYNCcnt path).
    for (int i = tid; i < cnt; i += 256) {
      unsigned lds = (unsigned)(uintptr_t)(&ebuf[i]);   // low 32b = LDS offset
      unsigned long long ga = (unsigned long long)(uintptr_t)(edata + cb + i);
      asm volatile("global_load_async_to_lds_b64 %0, %1, off"
                   :: "v"(lds), "v"(ga) : "memory");
    }
    asm volatile("s_wait_asynccnt 0" ::: "memory");
    __syncthreads();

    const int lo = (myBeg > cb) ? myBeg : cb;
    const int hi = (myEnd < cb + cnt) ? myEnd : (cb + cnt);
    for (int e = lo; e < hi; ++e) {
      const int2 p  = ebuf[e - cb];
      const float w = __int_as_float(p.y);
      const float4 hv = *reinterpret_cast<const float4*>(h + (size_t)p.x * F + fo);
      acc.x += w * hv.x; acc.y += w * hv.y; acc.z += w * hv.z; acc.w += w * hv.w;
    }
    __syncthreads();   // protect ebuf before next chunk overwrites it
  }

  if (node < n) {
    const float4 hd = *reinterpret_cast<const float4*>(h + (size_t)node * F + fo);
    float4 r;
    r.x = 0.5f * (hd.x + acc.x);
    r.y = 0.5f * (hd.y + acc.y);
    r.z = 0.5f * (hd.z + acc.z);
    r.w = 0.5f * (hd.w + acc.w);
    *reinterpret_cast<float4*>(hout + (size_t)node * F + fo) = r;
    if constexpr (F == 256) {
      if (out2 != nullptr) {
        const int k = fo >> 6, c0 = fo & 63;
        *reinterpret_cast<float4*>(
            out2 + ((size_t)node * 73 + (size_t)(5 + k * 17 + t)) * 64 + c0) = r;
      }
    }
  }
}

// ---------------------------------------------------------------------------
// First-cascade epilogue as a WMMA matmul per 16-node tile and channel:
//   D(16 nodes x 16 cols) = A(16 nodes x 4 states) x B(4 states x 16)
// A layout (ISA 7.12.2, 32-bit A 16x4): lane m<16 holds row m; VGPR0={K0,K2},
// VGPR1={K1,K3} across lane halves. B rows striped across lanes per VGPR:
// VGPR r = rows K=r (lanes 0-15) and K=r+2 (lanes 16-31). Columns j=0..3 of D
// give f1=a1-a2, f2=a2-a4, f3=a4-a8, f4=a8 (a16 subtracted at store, f32).
// Wave-uniform branch keeps EXEC all-ones for WMMA; scalar tail otherwise.
// ---------------------------------------------------------------------------
__global__ void __launch_bounds__(256)
combine_wmma_kernel(const float* __restrict__ x,
                    const float* __restrict__ a1, const float* __restrict__ a2,
                    const float* __restrict__ a4, const float* __restrict__ a8,
                    const float* __restrict__ a16,
                    float* __restrict__ out, float* __restrict__ s1, int n) {
  const int base = blockIdx.x * 16;
  if (base + 16 <= n) {
    const int lane = threadIdx.x & 31;
    const int wave = threadIdx.x >> 5;   // 8 waves -> 8 channels each
    const int half = lane >> 4;
    const int q    = lane & 15;          // A row (node) / D column index

    // Constant coefficient matrix B[k][j] (difference operator).
    v2f Bv;
    if (half == 0) {
      Bv.x = (q == 0) ? 1.f : 0.f;                        // K=0 row
      Bv.y = (q == 0) ? -1.f : ((q == 1) ? 1.f : 0.f);    // K=1 row
    } else {
      Bv.x = (q == 1) ? -1.f : ((q == 2) ? 1.f : 0.f);    // K=2 row
      Bv.y = (q == 2) ? -1.f : ((q == 3) ? 1.f : 0.f);    // K=3 row
    }

    for (int cc = 0; cc < 8; ++cc) {
      const int c = wave * 8 + cc;
      const size_t idx = (size_t)(base + q) * 64 + c;
      v2f Av;
      if (half == 0) { Av.x = a1[idx]; Av.y = a2[idx]; }   // K=0, K=1
      else           { Av.x = a4[idx]; Av.y = a8[idx]; }   // K=2, K=3
      v8f d = {};
      d = __builtin_amdgcn_wmma_f32_16x16x4_f32(
              /*neg_a=*/false, Av, /*neg_b=*/false, Bv,
              /*c_mod=*/(short)0, d, /*reuse_a=*/false, /*reuse_b=*/false);

      const int j = q;  // D column owned by this lane (lanes j and j+16)
      if (j < 4) {
#pragma unroll
        for (int r = 0; r < 8; ++r) {
          const int node = base + r + half * 8;            // D row M
          float val = d[r];
          if (j == 3) val -= a16[(size_t)node * 64 + c];   // f4 = a8 - a16
          val = fabsf(val);
          float* ob = out + (size_t)node * 73 * 64;
          ob[(1 + j) * 64 + c]      = val;                 // rows 1..4
          ob[(5 + j * 17) * 64 + c] = val;                 // cascade-2 t=0 rows
          s1[(size_t)node * 256 + j * 64 + c] = val;       // [N,4,64] k-major
        }
      } else if (j == 4) {
#pragma unroll
        for (int r = 0; r < 8; ++r) {                      // row 0: s0 = x
          const int node = base + r + half * 8;
          out[(size_t)node * 73 * 64 + c] = x[(size_t)node * 64 + c];
        }
      }
    }
  } else {
    // Scalar tail for a partial tile (keeps WMMA path EXEC-uniform).
    const int cnt = (n - base) * 64;
    for (int idx = threadIdx.x; idx < cnt; idx += 256) {
      int node = base + (idx >> 6), c = idx & 63;
      size_t i = (size_t)node * 64 + c;
      float f1 = fabsf(a1[i] - a2[i]);
      float f2 = fabsf(a2[i] - a4[i]);
      float f3 = fabsf(a4[i] - a8[i]);
      float f4 = fabsf(a8[i] - a16[i]);
      float* ob = out + (size_t)node * 73 * 64;
      ob[c]          = x[i];
      ob[64 + c]     = f1; ob[128 + c] = f2;
      ob[192 + c]    = f3; ob[256 + c] = f4;
      ob[(5 + 0 * 17) * 64 + c] = f1;
      ob[(5 + 1 * 17) * 64 + c] = f2;
      ob[(5 + 2 * 17) * 64 + c] = f3;
      ob[(5 + 3 * 17) * 64 + c] = f4;
      float* sb = s1 + (size_t)node * 256;
      sb[c] = f1; sb[64 + c] = f2; sb[128 + c] = f3; sb[192 + c] = f4;
    }
  }
}

extern "C" void kernel_launch(void* const* d_in, const int* in_sizes, int n_in,
                              void* d_out, int out_size, void* d_ws, size_t ws_size,
                              hipStream_t stream) {
  (void)n_in; (void)out_size; (void)ws_size;
  const int C = 64;
  const float* x  = (const float*)d_in[0];
  const int*   ei = (const int*)d_in[1];
  const int N = in_sizes[0] / C;
  const int E = in_sizes[1] / 2;
  const int* row = ei;
  const int* col = ei + E;
  float* out = (float*)d_out;

  // --- workspace layout (all 256B aligned) -------------------------------
  char* ws = (char*)d_ws;
  size_t off = 0;
  auto alloc = [&](size_t bytes) -> char* {
    char* p = ws + off;
    off = (off + bytes + 255) & ~(size_t)255;
    return p;
  };
  int*   deg   = (int*)  alloc((size_t)N * 4);
  int*   fill  = (int*)  alloc((size_t)N * 4);
  float* dinv  = (float*)alloc((size_t)N * 4);
  int*   csr   = (int*)  alloc((size_t)(N + 1) * 4);
  int2*  edata = (int2*) alloc((size_t)E * 8);
  char*  snapRegion = ws + off;                       // 4 snapshots = N*256*4 B
  float* snap1 = (float*)alloc((size_t)N * 64 * 4);
  float* snap2 = (float*)alloc((size_t)N * 64 * 4);
  float* snap4 = (float*)alloc((size_t)N * 64 * 4);
  float* snap8 = (float*)alloc((size_t)N * 64 * 4);
  float* tmpA  = (float*)alloc((size_t)N * 64 * 4);
  float* tmpB  = (float*)alloc((size_t)N * 64 * 4);
  float* s1    = (float*)alloc((size_t)N * 256 * 4);  // cascade-2 state A
  float* h2b   = (float*)snapRegion;                  // cascade-2 state B: reuses
                                                      // dead snapshot region

  const int gN = (N + 255) / 256;
  const int gE = (E + 255) / 256;

  // --- CSR build + normalization ----------------------------------------
  zero2_kernel  <<<gN, 256, 0, stream>>>(deg, fill, N);
  count_kernel  <<<gE, 256, 0, stream>>>(col, deg, E);
  dinv_kernel   <<<gN, 256, 0, stream>>>(deg, dinv, N);
  scan_kernel   <<<1, 1024, 0, stream>>>(deg, csr, N);
  scatter_kernel<<<gE, 256, 0, stream>>>(row, col, dinv, csr, fill, edata, E);

  // --- first cascade: 16 diffusions on [N,64]; snapshot t = 1,2,4,8 ------
  const float* cur = x;
  for (int t = 1; t <= 16; ++t) {
    float* dst;
    if      (t == 1) dst = snap1;
    else if (t == 2) dst = snap2;
    else if (t == 4) dst = snap4;
    else if (t == 8) dst = snap8;
    else             dst = (cur == tmpA) ? tmpB : tmpA;
    diffuse_kernel<64><<<(N + 15) / 16, 256, 0, stream>>>(
        cur, dst, csr, edata, N, nullptr, 0);
    cur = dst;
  }
  const float* a16 = cur;   // lands in tmpA/tmpB, not a snapshot

  combine_wmma_kernel<<<(N + 15) / 16, 256, 0, stream>>>(
      x, snap1, snap2, snap4, snap8, a16, out, s1, N);

  // --- second cascade: 16 diffusions on [N,256]; rows written to d_out ---
  float* c2 = s1;
  float* nx = h2b;   // snapshots are dead after combine; safe to reuse
  for (int t = 1; t <= 16; ++t) {
    diffuse_kernel<256><<<(N + 3) / 4, 256, 0, stream>>>(
        c2, nx, csr, edata, N, out, t);
    float* sw = c2; c2 = nx; nx = sw;
  }
}